// TripletLoss_margins_55972013802306
// MI455X (gfx1250) — compile-verified
//
#include <hip/hip_runtime.h>

typedef float v4f __attribute__((ext_vector_type(4)));

#define THREADS 256
#define BLOCKS  2048
#define WAVES_PER_BLOCK (THREADS / 32)
#define ILP 4

// ds_swizzle_b32 with immediate group-of-32 pattern: no lane-index ALU needed.
// offset[14:10]=xor_mask, offset[9:5]=or_mask, offset[4:0]=and_mask.
#define SWZ_X16 0x401F
#define SWZ_X8  0x201F
#define SWZ_X4  0x101F
#define SWZ_X2  0x081F
#define SWZ_X1  0x041F

template <int PATTERN>
__device__ __forceinline__ float swz(float x) {
    return __builtin_bit_cast(
        float, __builtin_amdgcn_ds_swizzle(__builtin_bit_cast(int, x), PATTERN));
}

__device__ __forceinline__ float wave32_reduce_add(float v) {
    v += swz<SWZ_X16>(v);
    v += swz<SWZ_X8>(v);
    v += swz<SWZ_X4>(v);
    v += swz<SWZ_X2>(v);
    v += swz<SWZ_X1>(v);
    return v;   // all 32 lanes hold the sum
}

__device__ __forceinline__ float dot4_sq(v4f d) {
    float s = d.x * d.x;
    s = fmaf(d.y, d.y, s);
    s = fmaf(d.z, d.z, s);
    s = fmaf(d.w, d.w, s);
    return s;
}

__device__ __forceinline__ float margin_of(int c) {
    // MARGINS = {0.1, 0.085, 0.07, 0.04}; select chain stays in VGPRs
    return (c == 0) ? 0.1f : (c == 1) ? 0.085f : (c == 2) ? 0.07f : 0.04f;
}

__global__ void __launch_bounds__(THREADS)
triplet_partials_kernel(const float* __restrict__ anchors,
                        const float* __restrict__ poss,
                        const float* __restrict__ negs,
                        const int*   __restrict__ cs,
                        float*       __restrict__ partials,
                        int B) {
    const int   lane    = threadIdx.x & 31;
    const int   waveBlk = threadIdx.x >> 5;
    const int   gwave   = blockIdx.x * WAVES_PER_BLOCK + waveBlk;
    const int   nwaves  = gridDim.x * WAVES_PER_BLOCK;
    const int   col     = lane * 4;     // one wave32 x float4 == one 128-float row
    const bool  lowHalf = lane < 16;
    const float eps     = 1e-6f;

    float acc = 0.0f;

    for (int row = gwave; row < B; row += ILP * nwaves) {
        int  r[ILP];
        bool ok[ILP];
        v4f  va[ILP], vp[ILP], vn[ILP];

        // Issue all loads first: 12 independent non-temporal b128 loads in
        // flight per wave (402MB streamed once > 192MB L2 -> NT hint).
        #pragma unroll
        for (int k = 0; k < ILP; ++k) {
            r[k]  = row + k * nwaves;
            ok[k] = (r[k] < B);                 // uniform per wave
            if (ok[k]) {
                const size_t base = (size_t)r[k] * 128 + col;
                va[k] = __builtin_nontemporal_load((const v4f*)(anchors + base));
                vp[k] = __builtin_nontemporal_load((const v4f*)(poss    + base));
                vn[k] = __builtin_nontemporal_load((const v4f*)(negs    + base));
                // Pre-stage next grid-stride chunk into GL2
                // (gfx1250: global_prefetch_b8; speculative => OOB-safe).
                const int rn = r[k] + ILP * nwaves;
                if (rn < B) {
                    const size_t nb = (size_t)rn * 128 + col;
                    __builtin_prefetch(anchors + nb, 0, 0);
                    __builtin_prefetch(poss    + nb, 0, 0);
                    __builtin_prefetch(negs    + nb, 0, 0);
                }
            }
        }

        #pragma unroll
        for (int k = 0; k < ILP; ++k) {
            if (!ok[k]) continue;
            v4f dap = va[k] - vp[k] + eps;      // (a - p + eps) per element
            v4f dan = va[k] - vn[k] + eps;
            float sap = dot4_sq(dap);
            float san = dot4_sq(dan);

            // Fused dual reduction: one cross-half swap puts the ap-tree in
            // lanes 0-15 and the an-tree in lanes 16-31, then both reduce in
            // the same 4 butterfly steps. 6 ds_swizzle + 1 sqrt per row.
            float mine  = lowHalf ? sap : san;
            float other = lowHalf ? san : sap;
            float v = mine + swz<SWZ_X16>(other);
            v += swz<SWZ_X8>(v);
            v += swz<SWZ_X4>(v);
            v += swz<SWZ_X2>(v);
            v += swz<SWZ_X1>(v);

            const float d  = sqrtf(v);          // d_ap (lanes 0-15) / d_an (16-31)
            const float do_ = swz<SWZ_X16>(d);  // the other distance
            const float diff = lowHalf ? (d - do_) : (do_ - d);   // d_ap - d_an
            const float m    = margin_of(cs[r[k]] & 3);
            acc += fmaxf(diff + m, 0.0f);       // same value in all lanes
        }
    }

    // Block reduction: one partial per block (fixed order => deterministic)
    __shared__ float sacc[WAVES_PER_BLOCK];
    if (lane == 0) sacc[waveBlk] = acc;
    __syncthreads();
    if (threadIdx.x == 0) {
        float t = 0.0f;
        #pragma unroll
        for (int i = 0; i < WAVES_PER_BLOCK; ++i) t += sacc[i];
        partials[blockIdx.x] = t;
    }
}

__global__ void __launch_bounds__(THREADS)
final_reduce_kernel(const float* __restrict__ partials, int nparts,
                    float* __restrict__ out, float invB) {
    float acc = 0.0f;
    for (int i = threadIdx.x; i < nparts; i += THREADS)
        acc += partials[i];
    acc = wave32_reduce_add(acc);

    __shared__ float sacc[WAVES_PER_BLOCK];
    const int lane = threadIdx.x & 31;
    const int w    = threadIdx.x >> 5;
    if (lane == 0) sacc[w] = acc;
    __syncthreads();
    if (threadIdx.x == 0) {
        float t = 0.0f;
        #pragma unroll
        for (int i = 0; i < WAVES_PER_BLOCK; ++i) t += sacc[i];
        out[0] = t * invB;  // mean; invB = 2^-18 is exact
    }
}

extern "C" void kernel_launch(void* const* d_in, const int* in_sizes, int n_in,
                              void* d_out, int out_size, void* d_ws, size_t ws_size,
                              hipStream_t stream) {
    const float* anchors = (const float*)d_in[0];
    const float* poss    = (const float*)d_in[1];
    const float* negs    = (const float*)d_in[2];
    const int*   cs      = (const int*)d_in[3];
    const int    B       = in_sizes[3];      // 262144

    float* partials = (float*)d_ws;          // BLOCKS * 4 bytes = 8 KB scratch

    triplet_partials_kernel<<<BLOCKS, THREADS, 0, stream>>>(
        anchors, poss, negs, cs, partials, B);
    final_reduce_kernel<<<1, THREADS, 0, stream>>>(
        partials, BLOCKS, (float*)d_out, 1.0f / (float)B);
}